// PointFeatureNet_9268539425516
// MI455X (gfx1250) — compile-verified
//
#include <hip/hip_runtime.h>
#include <hip/hip_bf16.h>
#include <math.h>

typedef __attribute__((ext_vector_type(2))) float    v2f;
typedef __attribute__((ext_vector_type(8))) float    v8f;
typedef __attribute__((ext_vector_type(4))) unsigned v4u;
typedef __attribute__((ext_vector_type(4))) int      v4i;
typedef __attribute__((ext_vector_type(8))) int      v8i;

#define MAX_OUT 102124
#define MAX_PAD 102128            // MAX_OUT rounded up to 16 rows (zero padded)
#define SR 480
#define SP 360
#define SZ 32
#define NBATCH 4
#define NCELL  (NBATCH*SR*SP*SZ)  // 22,118,400
#define NWORDS (NCELL/32)         // 691,200
#define NSB    (NWORDS/256)       // 2,700 scan blocks

// ---------------------------------------------------------------------------
// f32 WMMA helper: D = A(16x4) * B(4x16) + C, fp32 throughout (CDNA5 wave32)
// ---------------------------------------------------------------------------
__device__ __forceinline__ v8f wmma_f32_k4(v2f a, v2f b, v8f c) {
  // 8 args: (neg_a, A, neg_b, B, c_mod, C, reuse_a, reuse_b)
  return __builtin_amdgcn_wmma_f32_16x16x4_f32(false, a, false, b, (short)0, c,
                                               false, false);
}

// ---------------------------------------------------------------------------
// Stage 1: voxelize. Per point: cylindrical coords, voxel key, 9-dim feature,
// set presence bit in the 22.1M-cell bitmap (2.77 MB, L2-resident).
// ---------------------------------------------------------------------------
__global__ void k_voxelize(const float* __restrict__ pts,
                           const int* __restrict__ bidx,
                           const float* __restrict__ cfg,
                           int* __restrict__ lin, float* __restrict__ pf,
                           unsigned* __restrict__ flags, int N) {
  int n = blockIdx.x * 256 + threadIdx.x;
  if (n >= N) return;
  float x = pts[(size_t)n*4+0], y = pts[(size_t)n*4+1];
  float z = pts[(size_t)n*4+2], it = pts[(size_t)n*4+3];
  float rho = sqrtf(x*x + y*y);
  float phi = atan2f(y, x);
  float cyl[3] = {rho, phi, z};
  const int S[3] = {SR, SP, SZ};
  int co[3]; float rel[3];
#pragma unroll
  for (int d = 0; d < 3; ++d) {
    float lo = cfg[d], hi = cfg[3+d];
    float vs = (hi - lo) / (float)S[d];
    int c = (int)floorf((cyl[d] - lo) / vs);
    c = c < 0 ? 0 : (c > S[d]-1 ? S[d]-1 : c);
    co[d] = c;
    rel[d] = cyl[d] - (lo + ((float)c + 0.5f) * vs);
  }
  int b = bidx[n];
  int L = ((b*SR + co[0])*SP + co[1])*SZ + co[2];
  lin[n] = L;
  float* o = pf + (size_t)n*9;
  o[0]=rel[0]; o[1]=rel[1]; o[2]=rel[2];
  o[3]=cyl[0]; o[4]=cyl[1]; o[5]=cyl[2];
  o[6]=x;      o[7]=y;      o[8]=it;
  atomicOr(&flags[(unsigned)L >> 5], 1u << ((unsigned)L & 31u));
}

// ---------------------------------------------------------------------------
// Stage 2: prefix scan of per-word popcounts -> sorted-unique rank lookup.
// ---------------------------------------------------------------------------
__global__ void k_scan1(const unsigned* __restrict__ flags,
                        unsigned* __restrict__ offs,
                        unsigned* __restrict__ bsums) {
  __shared__ unsigned s[256];
  int tid = threadIdx.x;
  int w = blockIdx.x * 256 + tid;
  unsigned c = (unsigned)__popc(flags[w]);
  s[tid] = c; __syncthreads();
  for (int d = 1; d < 256; d <<= 1) {
    unsigned t = (tid >= d) ? s[tid - d] : 0u; __syncthreads();
    s[tid] += t; __syncthreads();
  }
  offs[w] = s[tid] - c;                 // exclusive within block
  if (tid == 255) bsums[blockIdx.x] = s[255];
}

__global__ void k_scan2(unsigned* __restrict__ bsums, int* __restrict__ numact) {
  __shared__ unsigned s[256];
  __shared__ unsigned carry;
  int tid = threadIdx.x;
  if (tid == 0) carry = 0u;
  __syncthreads();
  for (int base = 0; base < NSB; base += 256) {
    int i = base + tid;
    unsigned c = (i < NSB) ? bsums[i] : 0u;
    s[tid] = c; __syncthreads();
    for (int d = 1; d < 256; d <<= 1) {
      unsigned t = (tid >= d) ? s[tid - d] : 0u; __syncthreads();
      s[tid] += t; __syncthreads();
    }
    unsigned incl = s[tid];
    unsigned cur = carry;
    if (i < NSB) bsums[i] = cur + incl - c;   // exclusive block offset
    __syncthreads();
    if (tid == 255) carry += s[255];
    __syncthreads();
  }
  if (tid == 0) {
    unsigned tot = carry;
    *numact = (int)(tot < (unsigned)MAX_OUT ? tot : (unsigned)MAX_OUT);
  }
}

__global__ void k_scan3(unsigned* __restrict__ offs,
                        const unsigned* __restrict__ bsums) {
  int w = blockIdx.x * 256 + threadIdx.x;
  offs[w] += bsums[blockIdx.x];
}

// Per-point sorted-unique rank (== jnp.unique inverse index), clamped.
__global__ void k_rank(const int* __restrict__ lin,
                       const unsigned* __restrict__ flags,
                       const unsigned* __restrict__ offs,
                       int* __restrict__ rank, int N) {
  int n = blockIdx.x * 256 + threadIdx.x;
  if (n >= N) return;
  unsigned L = (unsigned)lin[n];
  unsigned w = L >> 5, b = L & 31u;
  unsigned r = offs[w] + (unsigned)__popc(flags[w] & ((1u << b) - 1u));
  rank[n] = (int)(r < (unsigned)MAX_OUT ? r : (unsigned)(MAX_OUT - 1));
}

// Emit out_coors for each unique key in sorted order.
__global__ void k_coors(const unsigned* __restrict__ flags,
                        const unsigned* __restrict__ offs,
                        int* __restrict__ coors) {
  int w = blockIdx.x * 256 + threadIdx.x;
  if (w >= NWORDS) return;
  unsigned bits = flags[w];
  unsigned base = offs[w];
  int i = 0;
  while (bits) {
    int b = __ffs(bits) - 1;
    bits &= bits - 1u;
    unsigned r = base + (unsigned)i; ++i;
    if (r < (unsigned)MAX_OUT) {
      unsigned key = ((unsigned)w << 5) + (unsigned)b;
      int oz = (int)(key % SZ); unsigned t = key / SZ;
      int op = (int)(t % SP);   t /= SP;
      int orh = (int)(t % SR);
      int ob  = (int)(t / SR);
      coors[(size_t)r*4+0] = ob;  coors[(size_t)r*4+1] = orh;
      coors[(size_t)r*4+2] = op;  coors[(size_t)r*4+3] = oz;
    }
  }
}

// ---------------------------------------------------------------------------
// Stage 3: p0 = relu(pf @ W0 + b0)  (K=9, pure VALU) + scatter-max into v0.
// 64 threads per point: one output channel each.
// ---------------------------------------------------------------------------
__global__ void k_pfn0_scatter(const float* __restrict__ pf,
                               const int* __restrict__ rank,
                               const float* __restrict__ W0,
                               const float* __restrict__ b0,
                               float* __restrict__ v0, int N) {
  int tid = threadIdx.x;
  int point = blockIdx.x * 4 + (tid >> 6);
  int c = tid & 63;
  if (point >= N) return;
  const float* p = pf + (size_t)point * 9;
  float acc = b0[c];
#pragma unroll
  for (int k = 0; k < 9; ++k) acc = fmaf(p[k], W0[k*64 + c], acc);
  acc = fmaxf(acc, 0.0f);
  // post-ReLU values are >= 0: uint-bit atomicMax is order-preserving.
  atomicMax((unsigned*)(v0 + (size_t)rank[point]*64 + c), __float_as_uint(acc));
}

// ---------------------------------------------------------------------------
// Stage 4: p1 = relu([p0 | v0[idx]] @ W1 + b1)  via V_WMMA_F32_16X16X4_F32,
// fused with scatter-max into v1. Block = 256 thr = 8 waves, 32 points,
// each wave owns one 16x16 output tile (2 row-tiles x 4 col-tiles).
//
// W1 (128x64 f32, 32 KB) is staged into LDS by the Tensor Data Mover with
// LDS padding (+8 DWORDs every 64) so rows land at a 72-float stride:
// two rows apart = 144 DWORDs = 16 banks -> the two half-wave B-fragment
// reads hit disjoint bank ranges (conflict-free). The copy overlaps the
// VALU p0-recompute into As; completion via s_wait_tensorcnt + barrier.
// ---------------------------------------------------------------------------
__global__ void k_pfn1_wmma(const float* __restrict__ pf,
                            const int* __restrict__ rank,
                            const float* __restrict__ W0,
                            const float* __restrict__ b0,
                            const float* __restrict__ v0,
                            const float* __restrict__ W1,
                            const float* __restrict__ b1,
                            float* __restrict__ v1, int N) {
  __shared__ float As[32][132];              // +4 pad: conflict-free column reads
  __shared__ float Ws[128][72];              // W1 staged by TDM, 72-float stride
  int tid = threadIdx.x;
  int t0 = blockIdx.x * 32;

  // --- Wave 0: issue TDM load of W1 -> Ws (EXEC ignored; one wave issues). ---
  if ((tid >> 5) == 0) {
    unsigned long long ga  = (unsigned long long)(const void*)W1;
    // Flat LDS addresses are {SHARED_BASE, offset[31:0]} -> low 32 bits are
    // the LDS byte offset the D# wants.
    unsigned lds = (unsigned)(unsigned long long)(void*)&Ws[0][0];
    // D# group 0: count=1 | lds_addr | global_addr(57b) | type=2
    v4u g0;
    g0.x = 1u;                                   // count=1, user mode
    g0.y = lds;                                  // lds_addr (bytes)
    g0.z = (unsigned)ga;                         // global_addr[31:0]
    g0.w = (unsigned)((ga >> 32) & 0x01FFFFFFu)  // global_addr[56:32]
         | (2u << 30);                           // type=2 ("image")
    // D# group 1: data_size=4B, pad_enable, pad every 64 DW by +8 DW,
    // tensor_dim0 = tile_dim0 = 8192 elements (1 line), tile_dim1 = 1.
    v8i g1;
    g1[0] = (int)((2u << 16) | (1u << 20) | (5u << 22) | (7u << 25));
    g1[1] = (int)((8192u & 0xFFFFu) << 16);      // tensor_dim0[15:0] @ bits63:48
    g1[2] = (int)(1u << 16);                     // tensor_dim0[31:16]=0, tensor_dim1=1
    g1[3] = (int)((8192u & 0xFFFFu) << 16);      // tile_dim0 @ bits127:112
    g1[4] = 1;                                   // tile_dim1=1, tile_dim2=0
    g1[5] = 8192;                                // tensor_dim0_stride[31:0]
    g1[6] = 0;
    g1[7] = 0;
    v4i gz4 = {0, 0, 0, 0};                      // groups 2/3: <=2-D tensor
    v8i gz8 = {0, 0, 0, 0, 0, 0, 0, 0};
    // clang-23 lane: 6-arg form (g0, g1, g2, g3, extra, cpol)
    __builtin_amdgcn_tensor_load_to_lds(g0, g1, gz4, gz4, gz8, 0);
    __builtin_amdgcn_s_wait_tensorcnt(0);
    // Prefetch next tile's point features while we're at it.
    __builtin_prefetch(pf + (size_t)(t0 + 32) * 9, 0, 1);
  }

  // Stage A rows: first 64 cols = recomputed p0 (cheap, K=9), next 64 = v0 gather.
  for (int i = tid; i < 32*64; i += 256) {
    int m = i >> 6, c = i & 63, point = t0 + m;
    float val = 0.0f;
    if (point < N) {
      const float* p = pf + (size_t)point * 9;
      float acc = b0[c];
#pragma unroll
      for (int k = 0; k < 9; ++k) acc = fmaf(p[k], W0[k*64 + c], acc);
      val = fmaxf(acc, 0.0f);
    }
    As[m][c] = val;
  }
  for (int i = tid; i < 32*64; i += 256) {
    int m = i >> 6, c = i & 63, point = t0 + m;
    As[m][64 + c] = (point < N) ? v0[(size_t)rank[point]*64 + c] : 0.0f;
  }
  __syncthreads();   // publishes both As (VALU) and Ws (TDM) to all 8 waves

  int wave = tid >> 5, lane = tid & 31;
  int pt = wave >> 2, ct = wave & 3;       // point-subtile, column-tile
  int ln = lane & 15, half = lane >> 4;
  int mb = pt * 16;

  v8f acc = {};
  for (int k = 0; k < 128; k += 4) {
    int kk = k + 2*half;                   // A 16x4 f32: vgpr0={K0|K2}, vgpr1={K1|K3}
    v2f a; a.x = As[mb + ln][kk];       a.y = As[mb + ln][kk + 1];
    v2f b; b.x = Ws[kk][ct*16 + ln];    b.y = Ws[kk + 1][ct*16 + ln];
    acc = wmma_f32_k4(a, b, acc);
  }

  float bias = b1[ct*16 + ln];
#pragma unroll
  for (int v = 0; v < 8; ++v) {            // C/D: row = v + 8*half, col = ln
    int lrow = mb + v + 8*half;
    int point = t0 + lrow;
    if (point < N) {
      float val = fmaxf(acc[v] + bias, 0.0f);
      atomicMax((unsigned*)(v1 + (size_t)rank[point]*64 + ct*16 + ln),
                __float_as_uint(val));
    }
  }
}

// ---------------------------------------------------------------------------
// Stage 5: voxel_feats = relu(v1 @ W2 + b2), 102124x64 @ 64x128, f32 WMMA.
// Block = 8 waves = 8 tiles; grid covers 6383 row-tiles x 8 col-tiles.
// v1 is padded to MAX_PAD zeroed rows, so A reads are unguarded.
// ---------------------------------------------------------------------------
__global__ void k_post_wmma(const float* __restrict__ v1,
                            const float* __restrict__ W2,
                            const float* __restrict__ b2,
                            float* __restrict__ out) {
  int tid = threadIdx.x;
  int wave = tid >> 5, lane = tid & 31;
  int rt = blockIdx.x;                 // row tile
  int ct = wave;                       // column tile 0..7
  int ln = lane & 15, half = lane >> 4;
  int m0 = rt * 16;

  v8f acc = {};
  for (int k = 0; k < 64; k += 4) {
    int kk = k + 2*half;
    v2f a; a.x = v1[(size_t)(m0 + ln)*64 + kk];
           a.y = v1[(size_t)(m0 + ln)*64 + kk + 1];
    v2f b; b.x = W2[kk*128 + ct*16 + ln];
           b.y = W2[(kk+1)*128 + ct*16 + ln];
    acc = wmma_f32_k4(a, b, acc);
  }

  float bias = b2[ct*16 + ln];
#pragma unroll
  for (int v = 0; v < 8; ++v) {
    int row = m0 + v + 8*half;
    if (row < MAX_OUT)
      out[(size_t)row*128 + ct*16 + ln] = fmaxf(acc[v] + bias, 0.0f);
  }
}

// ---------------------------------------------------------------------------
extern "C" void kernel_launch(void* const* d_in, const int* in_sizes, int n_in,
                              void* d_out, int out_size, void* d_ws, size_t ws_size,
                              hipStream_t stream) {
  const float* pts  = (const float*)d_in[0];
  const int*   bidx = (const int*)d_in[1];
  const float* cfg  = (const float*)d_in[2];
  // d_in[3] = in_spatial_shape (480,360,32) — compile-time constants here.
  const float* W0 = (const float*)d_in[4];
  const float* b0 = (const float*)d_in[5];
  const float* W1 = (const float*)d_in[6];
  const float* b1 = (const float*)d_in[7];
  const float* W2 = (const float*)d_in[8];
  const float* b2 = (const float*)d_in[9];
  int N = in_sizes[0] / 4;

  // Workspace carve-out (~80 MB total)
  char* w = (char*)d_ws;
  auto carve = [&](size_t bytes) {
    char* p = w; w += (bytes + 255) & ~(size_t)255; return p;
  };
  int*      lin   = (int*)     carve((size_t)N * 4);
  int*      rank  = (int*)     carve((size_t)N * 4);
  float*    pf    = (float*)   carve((size_t)N * 9 * 4);
  unsigned* flags = (unsigned*)carve((size_t)NWORDS * 4);
  unsigned* offs  = (unsigned*)carve((size_t)NWORDS * 4);
  unsigned* bsums = (unsigned*)carve((size_t)NSB * 4);
  float*    v0    = (float*)   carve((size_t)MAX_PAD * 64 * 4);
  float*    v1    = (float*)   carve((size_t)MAX_PAD * 64 * 4);

  float* vox    = (float*)d_out;                      // MAX_OUT x 128 f32
  int*   coors  = (int*)d_out + (size_t)MAX_OUT*128;  // MAX_OUT x 4 int32
  int*   numact = coors + (size_t)MAX_OUT*4;          // 1 int32

  (void)hipMemsetAsync(flags, 0, (size_t)NWORDS * 4, stream);
  (void)hipMemsetAsync(v0, 0, (size_t)MAX_PAD * 64 * 4, stream);
  (void)hipMemsetAsync(v1, 0, (size_t)MAX_PAD * 64 * 4, stream);
  (void)hipMemsetAsync(coors, 0, ((size_t)MAX_OUT*4 + 1) * 4, stream);

  int nb = (N + 255) / 256;
  k_voxelize<<<nb, 256, 0, stream>>>(pts, bidx, cfg, lin, pf, flags, N);
  k_scan1<<<NSB, 256, 0, stream>>>(flags, offs, bsums);
  k_scan2<<<1, 256, 0, stream>>>(bsums, numact);
  k_scan3<<<NSB, 256, 0, stream>>>(offs, bsums);
  k_rank<<<nb, 256, 0, stream>>>(lin, flags, offs, rank, N);
  k_coors<<<(NWORDS + 255) / 256, 256, 0, stream>>>(flags, offs, coors);
  k_pfn0_scatter<<<(N + 3) / 4, 256, 0, stream>>>(pf, rank, W0, b0, v0, N);
  k_pfn1_wmma<<<(N + 31) / 32, 256, 0, stream>>>(pf, rank, W0, b0, v0, W1, b1, v1, N);
  k_post_wmma<<<(MAX_OUT + 15) / 16, 256, 0, stream>>>(v1, W2, b2, vox);
}